// WpSumAgg_28295244546187
// MI455X (gfx1250) — compile-verified
//
#include <hip/hip_runtime.h>
#include <hip/hip_bf16.h>
#include <stdint.h>

// Problem constants (from reference)
#define B_    1
#define T_    4
#define C_    64
#define H_    256
#define W_    256
#define HD_   4
#define CH_   16      // C/HD
#define Q_    4096
#define K0_   15
#define KTOP_ 10
#define PS_   7
#define NPOS_ (PS_ * PS_)          // 49 patch positions
#define PATCH_ (CH_ * NPOS_)       // 784 elements, 3136 bytes

typedef unsigned int u32;
typedef unsigned long long u64;
typedef __attribute__((ext_vector_type(4))) u32 v4u;
typedef __attribute__((ext_vector_type(4))) int v4i;
typedef __attribute__((ext_vector_type(8))) int v8i;

// One wave32 per (h, q) pair.
//  - Issues KTOP_ TDM tile loads (7 x 7 x 16 f32 tile, strides 1 / 256 / 65536
//    elements) into LDS; each patch lands contiguous (x-fastest, 784 dwords).
//  - Pipelines consumption with s_wait_tensorcnt (TDM completes in-order).
//  - Lane owns patch positions rem = lane and rem = lane+32 (lane<17), and all
//    16 channels of each -> output written as 4x b128 per position (contiguous
//    64B per lane), instead of scattered b32 stores.
__global__ __launch_bounds__(32) void WpSumAgg_tdm_kernel(
    const float* __restrict__ vid,
    const float* __restrict__ dists,
    const int*   __restrict__ inds,
    float*       __restrict__ out)
{
    __shared__ float smem[KTOP_ * PATCH_];   // 31360 B

    const int lane = (int)(threadIdx.x & 31u);
    const u32 bid  = blockIdx.x;
    const u32 h    = bid >> 12;      // / Q_
    const u32 q    = bid & 4095u;    // % Q_

    const u32 dbase = (h * (u32)Q_ + q) * (u32)K0_;

    // Weights (top-10 of 15)
    float w[KTOP_];
#pragma unroll
    for (int k = 0; k < KTOP_; ++k) w[k] = dists[dbase + (u32)k];

    // --- Constant D# groups (wave-uniform) ---------------------------------
    // Group1: data_size=4B; tensor_dim0=W=256; tensor_dim1=H=256;
    //         tile_dim0=7 (x), tile_dim1=7 (y), tile_dim2=16 (c);
    //         dim0_stride=256 (row), dim1_stride=65536 (channel plane)
    const v8i g1 = { 0x00020000,          // data_size=2 (4B), no pad/iterate/multicast
                     0x01000000,          // tensor_dim0[15:0]=256 in [31:16]
                     0x01000000,          // tensor_dim1[15:0]=256 in [31:16]
                     0x00070000,          // tile_dim0=7 in [31:16]
                     0x00100007,          // tile_dim1=7, tile_dim2=16
                     256,                 // tensor_dim0_stride lo32
                     0,                   // dim0_stride hi16 | dim1_stride[15:0]
                     1 };                 // tensor_dim1_stride[47:16] = 65536>>16
    // Group2: tensor_dim2=16 (channel extent); no dim3
    const v4i g2 = { 16, 0, 0, 0 };
    const v4i g3 = { 0, 0, 0, 0 };

    const u32 lds_base = (u32)(u64)(uintptr_t)&smem[0];
    const u64 vid_addr = (u64)(uintptr_t)vid;

    // --- Issue all KTOP_ tensor loads (TENSORcnt tracks, completes in order)
#pragma unroll
    for (int k = 0; k < KTOP_; ++k) {
        const u32 ib = (dbase + (u32)k) * 3u;
        int t  = __builtin_amdgcn_readfirstlane(inds[ib + 0]);
        int ii = __builtin_amdgcn_readfirstlane(inds[ib + 1]);
        int jj = __builtin_amdgcn_readfirstlane(inds[ib + 2]);

        // element offset of patch origin: (t*64 + h*16)*65536 + ii*256 + jj
        const u64 ga = vid_addr +
            4ull * ((u64)((u32)t * 64u + h * 16u) * 65536ull +
                    (u64)(u32)ii * 256ull + (u64)(u32)jj);

        v4u g0;
        g0.x = 1u;                                   // count=1 (valid), normal mode
        g0.y = lds_base + (u32)k * (u32)(PATCH_ * 4);// lds_addr (bytes)
        g0.z = (u32)ga;                              // global_addr[31:0]
        g0.w = ((u32)(ga >> 32) & 0x01FFFFFFu)       // global_addr[56:32]
               | 0x80000000u;                        // type=2 ("image")
#if __clang_major__ >= 23
        const v8i z8 = {0, 0, 0, 0, 0, 0, 0, 0};
        __builtin_amdgcn_tensor_load_to_lds(g0, g1, g2, g3, z8, 0);
#else
        __builtin_amdgcn_tensor_load_to_lds(g0, g1, g2, g3, 0);
#endif
    }

    // --- Pipelined accumulate: wait for patch k only -----------------------
    // Lane owns positions rem0 = lane (all lanes) and rem1 = lane+32 (lane<17),
    // accumulating all 16 channels of each in registers.
    float a0[CH_], a1[CH_];
#pragma unroll
    for (int c = 0; c < CH_; ++c) { a0[c] = 0.0f; a1[c] = 0.0f; }

    const int rem0 = lane;
    const int rem1 = lane + 32;   // valid (<49) only for lane < 17; loads past
                                  // the patch are harmless, store is guarded.

#define WPSUM_STEP(k)                                                        \
    {                                                                        \
        __builtin_amdgcn_s_wait_tensorcnt((KTOP_ - 1) - (k));                \
        const float wk_ = w[(k)];                                            \
        const float* pk_ = &smem[(k) * PATCH_];                              \
        _Pragma("unroll")                                                    \
        for (int c = 0; c < CH_; ++c) {                                      \
            a0[c] = __builtin_fmaf(wk_, pk_[c * NPOS_ + rem0], a0[c]);       \
            a1[c] = __builtin_fmaf(wk_, pk_[c * NPOS_ + rem1], a1[c]);       \
        }                                                                    \
    }
    WPSUM_STEP(0) WPSUM_STEP(1) WPSUM_STEP(2) WPSUM_STEP(3) WPSUM_STEP(4)
    WPSUM_STEP(5) WPSUM_STEP(6) WPSUM_STEP(7) WPSUM_STEP(8) WPSUM_STEP(9)
#undef WPSUM_STEP

    // --- Store: out[(q*49 + rem)*64 + h*16 + c] as 4x float4 per position --
    const u32 ob = q * (u32)(NPOS_ * 64) + h * 16u;

    float4* o0 = (float4*)(out + ob + (u32)rem0 * 64u);
#pragma unroll
    for (int c4 = 0; c4 < 4; ++c4) {
        float4 v;
        v.x = a0[c4 * 4 + 0]; v.y = a0[c4 * 4 + 1];
        v.z = a0[c4 * 4 + 2]; v.w = a0[c4 * 4 + 3];
        o0[c4] = v;
    }
    if (lane < (NPOS_ - 32)) {   // lane < 17
        float4* o1 = (float4*)(out + ob + (u32)rem1 * 64u);
#pragma unroll
        for (int c4 = 0; c4 < 4; ++c4) {
            float4 v;
            v.x = a1[c4 * 4 + 0]; v.y = a1[c4 * 4 + 1];
            v.z = a1[c4 * 4 + 2]; v.w = a1[c4 * 4 + 3];
            o1[c4] = v;
        }
    }
}

extern "C" void kernel_launch(void* const* d_in, const int* in_sizes, int n_in,
                              void* d_out, int out_size, void* d_ws, size_t ws_size,
                              hipStream_t stream) {
    (void)in_sizes; (void)n_in; (void)out_size; (void)d_ws; (void)ws_size;
    const float* vid   = (const float*)d_in[0];
    const float* dists = (const float*)d_in[1];
    const int*   inds  = (const int*)d_in[2];
    float*       out   = (float*)d_out;

    dim3 grid(HD_ * Q_);   // 16384 workgroups, one wave32 each
    dim3 block(32);
    WpSumAgg_tdm_kernel<<<grid, block, 0, stream>>>(vid, dists, inds, out);
}